// BiasedDropout_38577396252773
// MI455X (gfx1250) — compile-verified
//
#include <hip/hip_runtime.h>

// ---------------------------------------------------------------------------
// Top-k row mask (torch.topk + scatter-back) for [B,N,C] fp32, C = 768.
// One wave32 per row. Exact radix-256 select (early-exit) on order-preserving
// integer keys, per-wave private LDS histograms (no barriers), wave32 shuffle
// suffix-scan. CDNA5 paths: global_load_async_to_lds_b128 double-buffered row
// staging (ASYNCcnt), s_wait_asynccnt/s_wait_dscnt, global_prefetch_b8,
// non-temporal stores.
// ---------------------------------------------------------------------------

#define C_DIM 768
#define VPL   24          // values per lane = 768 / 32

typedef float vfloat4 __attribute__((ext_vector_type(4)));  // native vector for NT stores

// order-preserving float->uint key: larger float <=> larger key
__device__ __forceinline__ unsigned f2key(unsigned b, unsigned flip) {
  unsigned m = (unsigned)(((int)b) >> 31) | 0x80000000u;
  return (b ^ m) ^ flip;
}
// inverse map back to the float bits
__device__ __forceinline__ float key2f(unsigned kb, unsigned flip) {
  kb ^= flip;
  unsigned b = (kb & 0x80000000u) ? (kb ^ 0x80000000u) : ~kb;
  return __uint_as_float(b);
}

// LDS byte offset of a generic pointer to __shared__ (flat addr[31:0] == LDS offset)
__device__ __forceinline__ unsigned lds_off(const void* p) {
  return (unsigned)(uintptr_t)p;
}

// Stage one 3KB row (768 fp32) from global into LDS with 6 async b128 ops.
// Lane j-th transfer covers bytes [(j*32+lane)*16, +16) of the row, matching
// the element layout c = j*128 + lane*4 + comp used by the compute phase.
__device__ __forceinline__ void async_row_load(const float* gbase,
                                               unsigned ldsBase, int lane) {
  #pragma unroll
  for (int j = 0; j < 6; ++j) {
    unsigned off = (unsigned)((j * 32 + lane) * 16);
    unsigned long long ga = (unsigned long long)(uintptr_t)gbase + off;
    asm volatile("global_load_async_to_lds_b128 %0, %1, off"
                 :: "v"(ldsBase + off), "v"(ga) : "memory");
  }
}

__global__ __launch_bounds__(256)
void topk_scatter_mask_kernel(const float* __restrict__ x,
                              float* __restrict__ out,
                              const int* __restrict__ kptr,
                              const int* __restrict__ largestptr,
                              int rows) {
  __shared__ unsigned histmem[8 * 256];      // 1KB of bins per wave
  __shared__ float    stage[8][2][C_DIM];    // 2 x 3KB staging per wave

  const int lane = threadIdx.x & 31;
  const int wid  = threadIdx.x >> 5;
  unsigned* h = &histmem[wid << 8];

  int k = *kptr;
  if (k > C_DIM) k = C_DIM;
  const unsigned flip = (*largestptr) ? 0u : 0xFFFFFFFFu;  // largest=0 -> reverse order

  const int wavesTotal = (int)((gridDim.x * blockDim.x) >> 5);
  const int wave0      = (int)((blockIdx.x * blockDim.x + threadIdx.x) >> 5);

  int buf = 0;
  if (wave0 < rows)
    async_row_load(x + (size_t)wave0 * C_DIM, lds_off(&stage[wid][0][0]), lane);

  for (int row = wave0; row < rows; row += wavesTotal) {
    const int nrow = row + wavesTotal;
    if (nrow < rows) {
      // stage next row while we compute on the current one
      async_row_load(x + (size_t)nrow * C_DIM, lds_off(&stage[wid][buf ^ 1][0]), lane);
      // warm GL2 two rows ahead
      int prow = row + 2 * wavesTotal;
      if (prow < rows) {
        const char* p = (const char*)(x + (size_t)prow * C_DIM) + lane * 96;
        __builtin_prefetch(p, 0, 1);
      }
      asm volatile("s_wait_asynccnt 0x6" ::: "memory");  // current row's 6 xfers done
    } else {
      asm volatile("s_wait_asynccnt 0x0" ::: "memory");
    }

    vfloat4* op = (vfloat4*)(out + (size_t)row * C_DIM);

    // ---- read 24 values/lane from LDS, convert to monotone keys
    const float4* sp = (const float4*)&stage[wid][buf][0];
    unsigned key[VPL];
    #pragma unroll
    for (int j = 0; j < 6; ++j) {
      float4 f = sp[j * 32 + lane];
      key[4 * j + 0] = f2key(__float_as_uint(f.x), flip);
      key[4 * j + 1] = f2key(__float_as_uint(f.y), flip);
      key[4 * j + 2] = f2key(__float_as_uint(f.z), flip);
      key[4 * j + 3] = f2key(__float_as_uint(f.w), flip);
    }

    if (k <= 0) {                    // degenerate: everything masked to zero
      vfloat4 z = {0.f, 0.f, 0.f, 0.f};
      #pragma unroll
      for (int j = 0; j < 6; ++j)
        __builtin_nontemporal_store(z, &op[lane + 32 * j]);
      buf ^= 1;
      continue;
    }

    // ---- radix-256 select for the k-th largest key (exact, early-exit)
    unsigned rem    = (unsigned)k;   // still needed within current prefix group
    unsigned prefix = 0;             // selected high bits (lower bits zero)
    unsigned ecnt   = 0;             // population of the selected bin

    #pragma unroll
    for (int p = 0; p < 4; ++p) {
      const int sh = 24 - 8 * p;

      // clear this wave's private 256 bins (8 per lane, contiguous)
      uint4 z4 = make_uint4(0u, 0u, 0u, 0u);
      ((uint4*)h)[lane * 2 + 0] = z4;
      ((uint4*)h)[lane * 2 + 1] = z4;

      // accumulate: keys matching the selected prefix participate.
      // match <=> top 8p bits equal <=> ((key^prefix)>>sh) < 256 (true for p==0)
      #pragma unroll
      for (int v = 0; v < VPL; ++v) {
        unsigned kb = key[v];
        if (((kb ^ prefix) >> sh) < 256u)
          atomicAdd(&h[(kb >> sh) & 255u], 1u);
      }

      asm volatile("s_wait_dscnt 0" ::: "memory");  // per-wave LDS ops in order

      uint4 b0 = ((uint4*)h)[lane * 2 + 0];   // bins 8*lane .. 8*lane+3
      uint4 b1 = ((uint4*)h)[lane * 2 + 1];   // bins 8*lane+4 .. 8*lane+7
      unsigned c[8] = {b0.x, b0.y, b0.z, b0.w, b1.x, b1.y, b1.z, b1.w};
      unsigned t = c[0] + c[1] + c[2] + c[3] + c[4] + c[5] + c[6] + c[7];

      // suffix sum over lanes (higher lane = higher bins), wave32
      unsigned S = t;
      #pragma unroll
      for (int off = 1; off < 32; off <<= 1) {
        unsigned v2 = __shfl_down(S, off, 32);
        S += (lane + off < 32) ? v2 : 0u;
      }
      unsigned above = S - t;                 // keys in bins above this lane's
      bool found = (S >= rem) && (above < rem);

      // locate threshold bin inside the found lane (scan own 8 bins top-down)
      int sb = 0; unsigned nr = 0, ec = 0;
      unsigned cum = above;
      #pragma unroll
      for (int i = 7; i >= 0; --i) {
        unsigned ci = c[i];
        if (found && nr == 0 && (cum + ci >= rem)) {
          sb = (lane << 3) | i;               // selected byte for this pass
          nr = rem - cum;                     // still needed inside that bin
          ec = ci;                            // population of that bin
        }
        cum += ci;
      }

      unsigned long long m = __ballot(found);
      int src = __ffsll(m) - 1;
      sb = __shfl(sb, src, 32);
      nr = __shfl((int)nr, src, 32);
      ec = __shfl((int)ec, src, 32);

      prefix |= ((unsigned)sb) << sh;
      rem  = nr;
      ecnt = ec;

      // early exit: whole bin selected -> key >= prefix (low bits 0) is exact
      if (ec == nr) break;
    }

    const unsigned thr = prefix;   // exact selection threshold
    const unsigned t   = rem;      // how many ==thr elements belong to the top-k

    if (ecnt == t) {
      // no ambiguous ties: keep everything >= threshold (exactly k elements)
      #pragma unroll
      for (int j = 0; j < 6; ++j) {
        vfloat4 o;
        o.x = (key[4 * j + 0] >= thr) ? key2f(key[4 * j + 0], flip) : 0.f;
        o.y = (key[4 * j + 1] >= thr) ? key2f(key[4 * j + 1], flip) : 0.f;
        o.z = (key[4 * j + 2] >= thr) ? key2f(key[4 * j + 2], flip) : 0.f;
        o.w = (key[4 * j + 3] >= thr) ? key2f(key[4 * j + 3], flip) : 0.f;
        __builtin_nontemporal_store(o, &op[lane + 32 * j]);
      }
    } else {
      // duplicated threshold value: top_k keeps the first t occurrences in
      // index order c = j*128 + lane*4 + comp. Ballot-based rank computation.
      const unsigned ltm = (1u << lane) - 1u;
      unsigned running = 0;   // equal-count at smaller j (wave-uniform)
      #pragma unroll
      for (int j = 0; j < 6; ++j) {
        unsigned e0 = (key[4 * j + 0] == thr) ? 1u : 0u;
        unsigned e1 = (key[4 * j + 1] == thr) ? 1u : 0u;
        unsigned e2 = (key[4 * j + 2] == thr) ? 1u : 0u;
        unsigned e3 = (key[4 * j + 3] == thr) ? 1u : 0u;
        unsigned bal0 = (unsigned)__ballot(e0 != 0);
        unsigned bal1 = (unsigned)__ballot(e1 != 0);
        unsigned bal2 = (unsigned)__ballot(e2 != 0);
        unsigned bal3 = (unsigned)__ballot(e3 != 0);
        unsigned lanePrev = __popc(bal0 & ltm) + __popc(bal1 & ltm) +
                            __popc(bal2 & ltm) + __popc(bal3 & ltm);
        unsigned base = running + lanePrev;
        bool s0 = e0 && (base < t);
        bool s1 = e1 && (base + e0 < t);
        bool s2 = e2 && (base + e0 + e1 < t);
        bool s3 = e3 && (base + e0 + e1 + e2 < t);
        vfloat4 o;
        o.x = (key[4 * j + 0] > thr || s0) ? key2f(key[4 * j + 0], flip) : 0.f;
        o.y = (key[4 * j + 1] > thr || s1) ? key2f(key[4 * j + 1], flip) : 0.f;
        o.z = (key[4 * j + 2] > thr || s2) ? key2f(key[4 * j + 2], flip) : 0.f;
        o.w = (key[4 * j + 3] > thr || s3) ? key2f(key[4 * j + 3], flip) : 0.f;
        __builtin_nontemporal_store(o, &op[lane + 32 * j]);
        running += __popc(bal0) + __popc(bal1) + __popc(bal2) + __popc(bal3);
      }
    }

    buf ^= 1;
  }
}

extern "C" void kernel_launch(void* const* d_in, const int* in_sizes, int n_in,
                              void* d_out, int out_size, void* d_ws, size_t ws_size,
                              hipStream_t stream) {
  const float* x       = (const float*)d_in[0];
  const int*   kp      = (const int*)d_in[1];
  const int*   largest = (const int*)d_in[2];
  float*       out     = (float*)d_out;

  const int rows = in_sizes[0] / C_DIM;          // B*N = 65536

  // one wave32 per row, ~4 rows per wave, 8 waves (256 threads) per block
  int waves  = (rows + 3) / 4;
  int blocks = (waves + 7) / 8;
  if (blocks < 1) blocks = 1;

  topk_scatter_mask_kernel<<<blocks, 256, 0, stream>>>(x, out, kp, largest, rows);
}